// E4_D4_32658931318953
// MI455X (gfx1250) — compile-verified
//
#include <hip/hip_runtime.h>

// MI455X (gfx1250) implementation.
// Dominant cost: two fp32 GEMMs (256x512x25088 and 512x512x25088) -> done with
// V_WMMA_F32_16X16X4_F32 (full fp32 precision vs the fp32 reference).
// GEMM uses double-buffered LDS fed by CDNA5 async global->LDS loads
// (GLOBAL_LOAD_ASYNC_TO_LDS_B128, ASYNCcnt) when available.

typedef __attribute__((ext_vector_type(2))) float v2f;
typedef __attribute__((ext_vector_type(8))) float v8f;
typedef int v4i __attribute__((vector_size(16)));

#define HW   3136
#define HH   56
#define WW   56
#define KDIM 512

#if defined(__HIP_DEVICE_COMPILE__) && __has_builtin(__builtin_amdgcn_global_load_async_to_lds_b128)
#define HAS_ASYNC 1
#else
#define HAS_ASYNC 0
#endif

#if HAS_ASYNC
#define ASYNC_B128(srcp, dstp)                                             \
  __builtin_amdgcn_global_load_async_to_lds_b128(                          \
      (__attribute__((address_space(1))) v4i*)(void*)(srcp),               \
      (__attribute__((address_space(3))) v4i*)(void*)(dstp), 0, 0)
__device__ __forceinline__ void wait_async0() {
  asm volatile("s_wait_asynccnt 0" ::: "memory");
}
#endif

// D4 regular-representation permutations (rows of PERMS in the reference)
__constant__ int d_perm[8][8] = {
  {0,1,2,3,4,5,6,7},
  {3,0,1,2,5,6,7,4},
  {2,3,0,1,6,7,4,5},
  {1,2,3,0,7,4,5,6},
  {4,5,6,7,0,1,2,3},
  {5,6,7,4,3,0,1,2},
  {6,7,4,5,2,3,0,1},
  {7,4,5,6,1,2,3,0},
};

// Tap-source table: trans[f][ti,tj] = s[f][a,b]; d_src[f][ti*3+tj] = a*3+b.
// Derived from the rot90/swapaxes chain in the reference:
//  f=0:(ti,tj) f=1:(tj,2-ti) f=2:(2-ti,2-tj) f=3:(2-tj,ti)
//  f=4:(ti,2-tj) f=5:(2-tj,2-ti) f=6:(2-ti,tj) f=7:(tj,ti)
__constant__ int d_src[8][9] = {
  {0,1,2,3,4,5,6,7,8},
  {2,5,8,1,4,7,0,3,6},
  {8,7,6,5,4,3,2,1,0},
  {6,3,0,7,4,1,8,5,2},
  {2,1,0,5,4,3,8,7,6},
  {8,5,2,7,4,1,6,3,0},
  {6,7,8,3,4,5,0,1,2},
  {0,3,6,1,4,7,2,5,8},
};

// ---------------------------------------------------------------------------
// Fold the D4 permutations into dense weight matrices:
//   Wfull[(o*8+g), (i*8+f)] = w[o, i, PERMS[g][f]]
// Wv: 512x512 rows (o<64), W1: 256x512 rows (o<32).
// ---------------------------------------------------------------------------
__global__ __launch_bounds__(256) void prep_kernel(
    const float* __restrict__ w_v, const float* __restrict__ w1,
    float* __restrict__ Wvf, float* __restrict__ W1f)
{
  int idx = blockIdx.x * 256 + threadIdx.x;            // < 393216
  if (idx < 262144) {
    int row = idx >> 9, col = idx & 511;
    int o = row >> 3, g = row & 7, i = col >> 3, f = col & 7;
    Wvf[idx] = w_v[((o << 6) + i) * 8 + d_perm[g][f]];
  } else {
    int j = idx - 262144;
    int row = j >> 9, col = j & 511;
    int o = row >> 3, g = row & 7, i = col >> 3, f = col & 7;
    W1f[j] = w1[((o << 6) + i) * 8 + d_perm[g][f]];
  }
}

// ---------------------------------------------------------------------------
// fp32 WMMA GEMM:  Y[b, n, m] = sum_k Wf[n,k] * x[b,k,m]
// Block tile: 128 out-channels x 64 spatial; K staged 16 at a time through a
// DOUBLE-BUFFERED LDS pipeline fed by async global->LDS B128 copies.
// 8 waves arranged 4(N) x 2(M); each wave owns a 32x32 output tile
// (2x2 of 16x16 f32 accumulators, 8 VGPRs each).
// A fragment (16x4 f32): lane L -> row L%16, K = 2*(L/16)+{0,1} (ISA layout).
// B fragment (4x16 f32): lane L -> col L%16, K = 2*(L/16)+{0,1} (mirror).
// D (16x16 f32): VGPR r, lane L -> row r+8*(L/16), col L%16.
// ---------------------------------------------------------------------------
__global__ __launch_bounds__(256) void gemm_kernel(
    const float* __restrict__ x, const float* __restrict__ Wf,
    float* __restrict__ Y, int nout)
{
  __shared__ float sA[2][128 * 20];   // [row][k], pitch 20 (conflict-free, 16B rows)
  __shared__ float sB[2][16 * 68];    // [k][m],  pitch 68 (16B rows)

  const int tid  = threadIdx.x;
  const int b    = blockIdx.z;
  const int nbase = blockIdx.y * 128;
  const int mbase = blockIdx.x * 64;
  const int wave = tid >> 5, lane = tid & 31;
  const int wn = wave & 3, wm = wave >> 2;
  const int half = lane >> 4, lr = lane & 15;

  // Per-thread staging coordinates.
  const int ar = tid >> 2;            // A row 0..63 (and +64 for second copy)
  const int ak = (tid & 3) * 4;       // A k-quad 0,4,8,12
  const int bkr = tid >> 4;           // B k-row 0..15
  const int bm  = (tid & 15) * 4;     // B m-quad

  const float* aSrc0 = Wf + (size_t)(nbase + ar) * KDIM + ak;
  const float* aSrc1 = aSrc0 + (size_t)64 * KDIM;
  const float* bSrc  = x + ((size_t)b * KDIM + bkr) * HW + mbase + bm;

  v8f acc[2][2];
  #pragma unroll
  for (int a = 0; a < 2; ++a)
    #pragma unroll
    for (int bb = 0; bb < 2; ++bb)
      #pragma unroll
      for (int r = 0; r < 8; ++r) acc[a][bb][r] = 0.0f;

  auto issue_stage = [&](int k0, int buf) {
#if HAS_ASYNC
    ASYNC_B128(aSrc0 + k0, &sA[buf][ar * 20 + ak]);
    ASYNC_B128(aSrc1 + k0, &sA[buf][(ar + 64) * 20 + ak]);
    ASYNC_B128(bSrc + (size_t)k0 * HW, &sB[buf][bkr * 68 + bm]);
#else
    float4 va0 = *(const float4*)(aSrc0 + k0);
    float4 va1 = *(const float4*)(aSrc1 + k0);
    float4 vb  = *(const float4*)(bSrc + (size_t)k0 * HW);
    *(float4*)&sA[buf][ar * 20 + ak] = va0;
    *(float4*)&sA[buf][(ar + 64) * 20 + ak] = va1;
    *(float4*)&sB[buf][bkr * 68 + bm] = vb;
#endif
  };

  issue_stage(0, 0);
#if HAS_ASYNC
  wait_async0();
#endif
  __syncthreads();

  const int NSTAGE = KDIM / 16;
  for (int s = 0; s < NSTAGE; ++s) {
    const int buf = s & 1;
    if (s + 1 < NSTAGE) issue_stage((s + 1) * 16, buf ^ 1);

    const float* cA = sA[buf];
    const float* cB = sB[buf];
    #pragma unroll
    for (int ks = 0; ks < 4; ++ks) {           // 4 x (K=4) sub-steps
      const int kk = ks * 4 + 2 * half;
      v2f afr[2], bfr[2];
      #pragma unroll
      for (int nt = 0; nt < 2; ++nt)
        afr[nt] = *(const v2f*)(cA + (wn * 32 + nt * 16 + lr) * 20 + kk);
      #pragma unroll
      for (int mt = 0; mt < 2; ++mt) {
        int col = wm * 32 + mt * 16 + lr;
        v2f bv; bv.x = cB[kk * 68 + col]; bv.y = cB[(kk + 1) * 68 + col];
        bfr[mt] = bv;
      }
      #pragma unroll
      for (int nt = 0; nt < 2; ++nt)
        #pragma unroll
        for (int mt = 0; mt < 2; ++mt)
          acc[nt][mt] = __builtin_amdgcn_wmma_f32_16x16x4_f32(
              false, afr[nt], false, bfr[mt], (short)0, acc[nt][mt], false, false);
    }

    // End-of-stage: my async loads for stage s+1 done, everyone's LDS reads of
    // buffer `buf` done (compiler waits DScnt before the barrier).
#if HAS_ASYNC
    wait_async0();
#endif
    __syncthreads();
  }

  // Epilogue: write 2x2 tiles; channel stride between D rows is HW.
  #pragma unroll
  for (int nt = 0; nt < 2; ++nt) {
    #pragma unroll
    for (int mt = 0; mt < 2; ++mt) {
      int n = nbase + wn * 32 + nt * 16 + 8 * half;
      int m = mbase + wm * 32 + mt * 16 + lr;
      float* yp = Y + ((size_t)b * nout + n) * HW + m;
      #pragma unroll
      for (int r = 0; r < 8; ++r) yp[(size_t)r * HW] = acc[nt][mt][r];
    }
  }
}

// ---------------------------------------------------------------------------
// Deterministic GroupNorm stats: one block per (b, mid-channel i); sums over
// 8 channels x 3136 pixels (contiguous 25088-float span).
// ---------------------------------------------------------------------------
__global__ __launch_bounds__(256) void reduce_stats_kernel(
    const float* __restrict__ t, float* __restrict__ stats)
{
  __shared__ float s1s[256], s2s[256];
  const int bi = blockIdx.x;                       // b*32 + i
  const float* p = t + (size_t)bi * 8 * HW;
  float s1 = 0.f, s2 = 0.f;
  for (int j = threadIdx.x; j < 8 * HW; j += 256) {
    float v = p[j]; s1 += v; s2 += v * v;
  }
  s1s[threadIdx.x] = s1; s2s[threadIdx.x] = s2;
  __syncthreads();
  for (int o = 128; o > 0; o >>= 1) {
    if (threadIdx.x < o) {
      s1s[threadIdx.x] += s1s[threadIdx.x + o];
      s2s[threadIdx.x] += s2s[threadIdx.x + o];
    }
    __syncthreads();
  }
  if (threadIdx.x == 0) { stats[bi * 2] = s1s[0]; stats[bi * 2 + 1] = s2s[0]; }
}

// Fold mean/var/gamma/beta into per-(b,i,f) affine (A, B): xn = A*x + B
__global__ void affine_kernel(const float* __restrict__ stats,
                              const float* __restrict__ gamma,
                              const float* __restrict__ beta,
                              float* __restrict__ aff)
{
  int tid = threadIdx.x;                           // b*32 + i
  int i = tid & 31;
  const float n = 8.0f * (float)HW;
  float mean = stats[tid * 2] / n;
  float var  = stats[tid * 2 + 1] / n - mean * mean;
  float inv  = rsqrtf(var + 1e-5f);
  #pragma unroll
  for (int f = 0; f < 8; ++f) {
    float A  = inv * gamma[i * 8 + f];
    float Bv = beta[i * 8 + f] - mean * A;
    aff[(tid * 8 + f) * 2]     = A;
    aff[(tid * 8 + f) * 2 + 1] = Bv;
  }
}

// ---------------------------------------------------------------------------
// wgt[b, o(36), f, m] = b2[o] + sum_i w2[o,i] * relu(A[b,i,f]*t[b,i*8+f,m] + B)
// ---------------------------------------------------------------------------
__global__ __launch_bounds__(256) void wgt_kernel(
    const float* __restrict__ t, const float* __restrict__ aff,
    const float* __restrict__ w2, const float* __restrict__ b2,
    float* __restrict__ wgt)
{
  __shared__ float w2s[36 * 32];
  __shared__ float b2s[36];
  __shared__ float affs[64];
  const int tid = threadIdx.x;
  const int f = blockIdx.y, b = blockIdx.z;
  for (int j = tid; j < 36 * 32; j += 256) w2s[j] = w2[j];
  if (tid < 36) b2s[tid] = b2[tid];
  if (tid < 64) {
    int i = tid >> 1;
    affs[tid] = aff[(((size_t)b * 32 + i) * 8 + f) * 2 + (tid & 1)];
  }
  __syncthreads();

  int m = blockIdx.x * 256 + tid;
  if (m >= HW) return;

  float acc[36];
  #pragma unroll
  for (int o = 0; o < 36; ++o) acc[o] = b2s[o];

  const float* tp = t + ((size_t)b * 256 + f) * HW + m;   // + i*8*HW per i
  for (int i = 0; i < 32; ++i) {
    float v = tp[(size_t)i * 8 * HW];
    v = fmaf(v, affs[2 * i], affs[2 * i + 1]);
    v = fmaxf(v, 0.0f);
    #pragma unroll
    for (int o = 0; o < 36; ++o) acc[o] = fmaf(w2s[o * 32 + i], v, acc[o]);
  }
  float* wp = wgt + (((size_t)b * 36) * 8 + f) * HW + m;  // + o*8*HW per o
  #pragma unroll
  for (int o = 0; o < 36; ++o) wp[(size_t)o * 8 * HW] = acc[o];
}

// ---------------------------------------------------------------------------
// Final dynamic depthwise 3x3 with D4 tap permutation:
// out[b, C, y, x] = sum_t wgt[b, Gi*9 + d_src[f][t], f, y, x] * xv[b, C, y+ti-1, x+tj-1]
// where C = ((Gi*16+c)*8+f). Block = (b, Gi, f, 4-row band); loop c (wgt reuse x16).
// ---------------------------------------------------------------------------
__global__ __launch_bounds__(224) void final_kernel(
    const float* __restrict__ wgt, const float* __restrict__ xv,
    float* __restrict__ out)
{
  __shared__ float wgtL[9][224];       // 9 taps x (4 rows x 56)
  __shared__ float xvL[6 * 58];        // (4+2 halo rows) x (56+2 halo cols)
  const int tid = threadIdx.x;         // 224 = 4 rows x 56
  const int y0 = blockIdx.x * 4;
  const int Gi = blockIdx.y >> 3, f = blockIdx.y & 7;
  const int b  = blockIdx.z;

  #pragma unroll
  for (int o = 0; o < 9; ++o)
    wgtL[o][tid] = wgt[(((size_t)b * 36 + Gi * 9 + o) * 8 + f) * HW + y0 * WW + tid];

  int src[9];
  #pragma unroll
  for (int t = 0; t < 9; ++t) src[t] = d_src[f][t];

  const int ry = tid / 56, xx = tid % 56;

  for (int c = 0; c < 16; ++c) {
    const int C = ((Gi * 16 + c) << 3) + f;
    const float* xvp = xv + ((size_t)b * 512 + C) * HW;
    __syncthreads();                       // also orders wgtL writes before reads
    for (int j = tid; j < 6 * 58; j += 224) {
      int r = j / 58, cc = j % 58;
      int gy = y0 + r - 1, gx = cc - 1;
      xvL[j] = (gy >= 0 && gy < HH && gx >= 0 && gx < WW) ? xvp[gy * WW + gx] : 0.0f;
    }
    __syncthreads();
    float s = 0.0f;
    #pragma unroll
    for (int t = 0; t < 9; ++t) {
      int ti = t / 3, tj = t % 3;
      s = fmaf(wgtL[src[t]][tid], xvL[(ry + ti) * 58 + xx + tj], s);
    }
    out[((size_t)b * 512 + C) * HW + y0 * WW + tid] = s;
  }
}

// ---------------------------------------------------------------------------
extern "C" void kernel_launch(void* const* d_in, const int* in_sizes, int n_in,
                              void* d_out, int out_size, void* d_ws, size_t ws_size,
                              hipStream_t stream)
{
  (void)in_sizes; (void)n_in; (void)out_size; (void)ws_size;
  const float* x     = (const float*)d_in[0];
  const float* w_v   = (const float*)d_in[1];
  const float* w1    = (const float*)d_in[2];
  const float* gamma = (const float*)d_in[3];
  const float* beta  = (const float*)d_in[4];
  const float* w2    = (const float*)d_in[5];
  const float* b2    = (const float*)d_in[6];
  float* out = (float*)d_out;
  float* ws  = (float*)d_ws;

  // Workspace layout (floats): total ~26.9M floats (~103 MB)
  float* Wvf   = ws;                         // 512*512        = 262144
  float* W1f   = Wvf   + 262144;             // 256*512        = 131072
  float* stats = W1f   + 131072;             // 8*32*2         = 512
  float* aff   = stats + 512;                // 8*256*2        = 4096
  float* t_raw = aff   + 4096;               // 8*256*3136     = 6422528
  float* wgt   = t_raw + 6422528;            // 8*288*3136     = 7225344
  float* xv    = wgt   + 7225344;            // 8*512*3136     = 12845056

  prep_kernel<<<dim3(1536, 1, 1), 256, 0, stream>>>(w_v, w1, Wvf, W1f);
  gemm_kernel<<<dim3(49, 2, 8), 256, 0, stream>>>(x, W1f, t_raw, 256);
  reduce_stats_kernel<<<dim3(256, 1, 1), 256, 0, stream>>>(t_raw, stats);
  affine_kernel<<<dim3(1, 1, 1), 256, 0, stream>>>(stats, gamma, beta, aff);
  wgt_kernel<<<dim3(13, 8, 8), 256, 0, stream>>>(t_raw, aff, w2, b2, wgt);
  gemm_kernel<<<dim3(49, 4, 8), 256, 0, stream>>>(x, Wvf, xv, 512);
  final_kernel<<<dim3(14, 32, 8), 224, 0, stream>>>(wgt, xv, out);
}